// SAGE_42812234006571
// MI455X (gfx1250) — compile-verified
//
#include <hip/hip_runtime.h>

typedef float v2f __attribute__((ext_vector_type(2)));
typedef float v8f __attribute__((ext_vector_type(8)));

#define D_FEAT 256
#define N1_NODES 40000
#define N2_NODES 10000

// ---------------------------------------------------------------------------
// Zero a float region (grid-stride not needed; exact-size launch)
// ---------------------------------------------------------------------------
__global__ void zero_kernel(float* __restrict__ p, long n) {
    long i = (long)blockIdx.x * blockDim.x + threadIdx.x;
    if (i < n) p[i] = 0.0f;
}

// ---------------------------------------------------------------------------
// Edge-parallel mean-aggregation numerator + degree count.
// 64 threads per edge, each thread moves 4 consecutive floats (b128 load,
// 4x global_atomic_add_f32 into an L2-resident accumulation buffer).
// ---------------------------------------------------------------------------
__global__ void scatter_add_kernel(const float* __restrict__ xsrc,
                                   const int*   __restrict__ src,
                                   const int*   __restrict__ dst,
                                   float*       __restrict__ agg,
                                   float*       __restrict__ cnt,
                                   int E) {
    long t = (long)blockIdx.x * blockDim.x + threadIdx.x;
    int edge = (int)(t >> 6);   // 64 threads per edge
    int lane = (int)(t & 63);   // 4 floats each -> 256 features
    if (edge >= E) return;
    int s = src[edge];
    int d = dst[edge];
    const float4 v = *reinterpret_cast<const float4*>(xsrc + (long)s * D_FEAT + lane * 4);
    float* ap = agg + (long)d * D_FEAT + lane * 4;
    atomicAdd(ap + 0, v.x);
    atomicAdd(ap + 1, v.y);
    atomicAdd(ap + 2, v.z);
    atomicAdd(ap + 3, v.w);
    if (lane == 0) atomicAdd(cnt + d, 1.0f);
}

// ---------------------------------------------------------------------------
// Fused: out[rows] = act( (agg/max(cnt,1)) @ Wl + bl + xt @ Wr )
// Block = 512 threads = 16 waves. Block handles 16 rows; wave w handles
// output columns [16w, 16w+16). A-tiles staged in LDS (scaled), weights
// streamed from global (L2-broadcast). Core op: V_WMMA_F32_16X16X4_F32.
// ---------------------------------------------------------------------------
template <bool RELU>
__global__ void sage_gemm_kernel(const float* __restrict__ agg,
                                 const float* __restrict__ cnt,
                                 const float* __restrict__ xt,
                                 const float* __restrict__ Wl,
                                 const float* __restrict__ bl,
                                 const float* __restrict__ Wr,
                                 float*       __restrict__ out) {
    constexpr int D = D_FEAT;
    constexpr int LDSS = D + 4;            // 260: conflict-free LDS stride
    __shared__ float As1[16][LDSS];        // scaled neighbor mean
    __shared__ float As2[16][LDSS];        // target features

    const int tid = threadIdx.x;
    const int rowbase = blockIdx.x * 16;

    // Cooperative LDS fill: 512 threads x 2 iters x float4 = 16x256 per matrix
#pragma unroll
    for (int i = 0; i < 2; ++i) {
        int lin = tid + i * 512;           // 0..1023
        int r   = lin >> 6;                // 0..15
        int c4  = (lin & 63) << 2;         // 0..252 step 4
        long g  = (long)(rowbase + r) * D + c4;
        float deg = cnt[rowbase + r];
        float inv = 1.0f / fmaxf(deg, 1.0f);
        float4 a = *reinterpret_cast<const float4*>(agg + g);
        As1[r][c4 + 0] = a.x * inv;
        As1[r][c4 + 1] = a.y * inv;
        As1[r][c4 + 2] = a.z * inv;
        As1[r][c4 + 3] = a.w * inv;
        float4 b = *reinterpret_cast<const float4*>(xt + g);
        As2[r][c4 + 0] = b.x;
        As2[r][c4 + 1] = b.y;
        As2[r][c4 + 2] = b.z;
        As2[r][c4 + 3] = b.w;
    }
    __syncthreads();

    const int wid  = tid >> 5;             // wave -> column tile
    const int L    = tid & 31;             // lane
    const int li   = L & 15;
    const int half = L >> 4;
    const int n    = wid * 16 + li;        // output column

    // Two accumulators to break the WMMA RAW chain; bias folded into c1.
    v8f c1, c2;
    float bias = bl[n];
#pragma unroll
    for (int r = 0; r < 8; ++r) { c1[r] = bias; c2[r] = 0.0f; }

#pragma unroll 2
    for (int kk = 0; kk < D; kk += 4) {
        const int k0 = kk + 2 * half;      // this lane's K pair
        v2f a1, b1, a2, b2;
        a1.x = As1[li][k0];
        a1.y = As1[li][k0 + 1];
        b1.x = Wl[(long)k0 * D + n];
        b1.y = Wl[(long)(k0 + 1) * D + n];
        c1 = __builtin_amdgcn_wmma_f32_16x16x4_f32(false, a1, false, b1,
                                                   (short)0, c1, false, false);
        a2.x = As2[li][k0];
        a2.y = As2[li][k0 + 1];
        b2.x = Wr[(long)k0 * D + n];
        b2.y = Wr[(long)(k0 + 1) * D + n];
        c2 = __builtin_amdgcn_wmma_f32_16x16x4_f32(false, a2, false, b2,
                                                   (short)0, c2, false, false);
    }

#pragma unroll
    for (int r = 0; r < 8; ++r) {
        float v = c1[r] + c2[r];
        if (RELU) v = fmaxf(v, 0.0f);
        out[(long)(rowbase + r + 8 * half) * D + n] = v;
    }
}

// ---------------------------------------------------------------------------
// Launch
// ---------------------------------------------------------------------------
extern "C" void kernel_launch(void* const* d_in, const int* in_sizes, int n_in,
                              void* d_out, int out_size, void* d_ws, size_t ws_size,
                              hipStream_t stream) {
    const float* x   = (const float*)d_in[0];
    const int*   src0 = (const int*)d_in[1];
    const int*   dst0 = (const int*)d_in[2];
    const int*   src1 = (const int*)d_in[3];
    const int*   dst1 = (const int*)d_in[4];
    const float* W0l = (const float*)d_in[5];
    const float* b0  = (const float*)d_in[6];
    const float* W0r = (const float*)d_in[7];
    const float* W1l = (const float*)d_in[8];
    const float* b1  = (const float*)d_in[9];
    const float* W1r = (const float*)d_in[10];
    float* out = (float*)d_out;

    const int E0 = in_sizes[1];
    const int E1 = in_sizes[3];

    // Workspace: [ h : N1*D f32 ][ agg : N1*D f32 ][ cnt : N1 f32 ]
    char*  ws  = (char*)d_ws;
    float* h   = (float*)ws;
    float* agg = (float*)(ws + (size_t)N1_NODES * D_FEAT * sizeof(float));
    float* cnt1 = agg + (size_t)N1_NODES * D_FEAT;   // hop-1 degree counts
    float* cnt2 = agg + (size_t)N2_NODES * D_FEAT;   // hop-2 (agg+cnt contiguous)

    // ---- hop 1: (N0 src, N1 tgt), relu ----
    {
        long nz = (long)N1_NODES * D_FEAT + N1_NODES;
        zero_kernel<<<(unsigned)((nz + 255) / 256), 256, 0, stream>>>(agg, nz);
        long nt = (long)E0 * 64;
        scatter_add_kernel<<<(unsigned)((nt + 255) / 256), 256, 0, stream>>>(
            x, src0, dst0, agg, cnt1, E0);
        sage_gemm_kernel<true><<<N1_NODES / 16, 512, 0, stream>>>(
            agg, cnt1, x, W0l, b0, W0r, h);
    }

    // ---- hop 2: (N1 src, N2 tgt), no activation ----
    {
        long nz = (long)N2_NODES * D_FEAT + N2_NODES;
        zero_kernel<<<(unsigned)((nz + 255) / 256), 256, 0, stream>>>(agg, nz);
        long nt = (long)E1 * 64;
        scatter_add_kernel<<<(unsigned)((nt + 255) / 256), 256, 0, stream>>>(
            h, src1, dst1, agg, cnt2, E1);
        sage_gemm_kernel<false><<<N2_NODES / 16, 512, 0, stream>>>(
            agg, cnt2, h, W1l, b1, W1r, out);
    }
}